// RZLayer_20624432956333
// MI455X (gfx1250) — compile-verified
//
#include <hip/hip_runtime.h>
#include <math.h>

// ---------------------------------------------------------------------------
// RZ diagonal-phase layer:  out[d,b] = e^{i*phi_d} * (xr[d,b] + i*xi[d,b])
// phi_d = sum_w (1 - 2*bit_w(d)) * theta_w / 2
// Memory bound: 512 MiB of traffic, ~23 us floor at 23.3 TB/s.
// ---------------------------------------------------------------------------

typedef float v4f __attribute__((ext_vector_type(4)));   // native vector: OK for
                                                         // nontemporal builtins
#define AS3 __attribute__((address_space(3)))

// Raw LDS byte offset of a __shared__ object (what the async-to-LDS VDST
// operand expects), via explicit generic -> addrspace(3) cast + ptrtoint.
__device__ __forceinline__ unsigned lds_addr_of(const void* p) {
    return (unsigned)(unsigned long long)(AS3 const char*)p;
}

// ---- Kernel 1: per-row phase table (cos/sin) into workspace ---------------
__global__ __launch_bounds__(256) void rz_phase_kernel(
    const float* __restrict__ theta,
    float* __restrict__ cs,
    float* __restrict__ sn,
    int n_wires, int dim)
{
    int d = blockIdx.x * 256 + threadIdx.x;
    if (d >= dim) return;
    float phase = 0.0f;
    for (int w = 0; w < n_wires; ++w) {
        // wire 0 is the MOST significant bit of the basis index (kron order)
        float sgn = ((d >> (n_wires - 1 - w)) & 1) ? -0.5f : 0.5f;
        phase = fmaf(sgn, theta[w], phase);
    }
    float s, c;
    sincosf(phase, &s, &c);
    cs[d] = c;
    sn[d] = s;
}

// ---- Kernel 2: streaming apply with double-buffered async-to-LDS ----------
// One block per row (dim=4096 rows). 256 threads * 4 elems = 1024-element
// tiles; batch/1024 tiles per block, double-buffered via ASYNCcnt.
__global__ __launch_bounds__(256) void rz_apply_kernel(
    const float* __restrict__ xr,
    const float* __restrict__ xi,
    const float* __restrict__ cs,
    const float* __restrict__ sn,
    float* __restrict__ out,
    int batch)
{
    __shared__ v4f s_r[2][256];
    __shared__ v4f s_i[2][256];

    const int row = blockIdx.x;
    const int t   = threadIdx.x;

    const float c = cs[row];
    const float s = sn[row];

    const float* rowr = xr  + (size_t)row * (size_t)batch;
    const float* rowi = xi  + (size_t)row * (size_t)batch;
    float*       rowo = out + (size_t)row * (size_t)batch * 2u;

    const unsigned lr0 = lds_addr_of(&s_r[0][t]);
    const unsigned lr1 = lds_addr_of(&s_r[1][t]);
    const unsigned li0 = lds_addr_of(&s_i[0][t]);
    const unsigned li1 = lds_addr_of(&s_i[1][t]);

    const int tiles = batch >> 10;     // 1024 elements per tile

    // Prologue: stage tile 0 into buffer 0 (2 async b128 loads per lane).
    {
        const float* ar = rowr + t * 4;
        const float* ai = rowi + t * 4;
        asm volatile(
            "global_load_async_to_lds_b128 %0, %1, off\n\t"
            "global_load_async_to_lds_b128 %2, %3, off"
            :: "v"(lr0), "v"(ar), "v"(li0), "v"(ai)
            : "memory");
    }

    for (int tile = 0; tile < tiles; ++tile) {
        const int buf = tile & 1;

        if (tile + 1 < tiles) {
            // Stage next tile into the other buffer, then wait until only
            // those 2 async ops remain (async loads retire in order).
            const float* ar = rowr + (size_t)(tile + 1) * 1024 + t * 4;
            const float* ai = rowi + (size_t)(tile + 1) * 1024 + t * 4;
            const unsigned lr = buf ? lr0 : lr1;
            const unsigned li = buf ? li0 : li1;
            asm volatile(
                "global_load_async_to_lds_b128 %0, %1, off\n\t"
                "global_load_async_to_lds_b128 %2, %3, off"
                :: "v"(lr), "v"(ar), "v"(li), "v"(ai)
                : "memory");
            asm volatile("s_wait_asynccnt 2" ::: "memory");
        } else {
            asm volatile("s_wait_asynccnt 0" ::: "memory");
        }

        // Each lane consumes exactly the 16B (x2) it staged: no barrier needed.
        const v4f r  = s_r[buf][t];
        const v4f im = s_i[buf][t];

        v4f o0, o1;
        o0.x = c * r.x - s * im.x;  o0.y = c * im.x + s * r.x;
        o0.z = c * r.y - s * im.y;  o0.w = c * im.y + s * r.y;
        o1.x = c * r.z - s * im.z;  o1.y = c * im.z + s * r.z;
        o1.z = c * r.w - s * im.w;  o1.w = c * im.w + s * r.w;

        // Interleaved complex64 output, write-once: non-temporal b128 stores.
        v4f* dst = (v4f*)(rowo + (size_t)(tile * 1024 + t * 4) * 2u);
        __builtin_nontemporal_store(o0, dst + 0);
        __builtin_nontemporal_store(o1, dst + 1);
    }
}

// ---------------------------------------------------------------------------
extern "C" void kernel_launch(void* const* d_in, const int* in_sizes, int n_in,
                              void* d_out, int out_size, void* d_ws, size_t ws_size,
                              hipStream_t stream)
{
    const float* theta = (const float*)d_in[0];   // [n_wires, 1]
    const float* xr    = (const float*)d_in[1];   // [dim, batch]
    const float* xi    = (const float*)d_in[2];   // [dim, batch]
    float*       out   = (float*)d_out;           // [dim, batch] complex64 (interleaved)

    const int n_wires = in_sizes[0];              // 12
    const int dim     = 1 << n_wires;             // 4096
    const int batch   = in_sizes[1] / dim;        // 8192

    float* cs = (float*)d_ws;                     // [dim]
    float* sn = cs + dim;                         // [dim]

    rz_phase_kernel<<<(dim + 255) / 256, 256, 0, stream>>>(theta, cs, sn, n_wires, dim);
    rz_apply_kernel<<<dim, 256, 0, stream>>>(xr, xi, cs, sn, out, batch);
}